// GAT_32315333935195
// MI455X (gfx1250) — compile-verified
//
#include <hip/hip_runtime.h>
#include <math.h>

#ifndef __has_builtin
#define __has_builtin(x) 0
#endif

#if defined(__gfx1250__) && __has_builtin(__builtin_amdgcn_wmma_f32_16x16x4_f32)
#define USE_WMMA_F32 1
#else
#define USE_WMMA_F32 0
#endif

typedef __attribute__((ext_vector_type(2))) float v2f;
typedef __attribute__((ext_vector_type(8))) float v8f;

// ---------------------------------------------------------------------------
// GEMM: C[M,N] = A[M,K] @ B[K,N], fp32, via V_WMMA_F32_16X16X4_F32.
// K, N are compile-time so all B strides fold into immediate offsets.
// One wave computes a 32(M) x 32(N) tile = 4 accumulators; A fragments are
// reused across the two column tiles, B fragments across the two row tiles
// -> 6 loads per 4 WMMAs. M%16==0 required; odd 16-row remainder tile clamps
// its second A fragment (wave-uniform, EXEC all-ones through WMMA) and skips
// the second store.
// VGPR layouts per CDNA5 ISA 7.12.2:
//   A 16x4 (MxK): lanes0-15 M=0..15; v0 = K=0 (lo half)/K=2 (hi half), v1 = K=1/K=3
//   B 4x16 (KxN): row striped across lanes; v0 = rows K=0 (lo)/K=2 (hi), v1 = K=1/K=3
//   C/D 16x16:    vgpr v: lanes0-15 -> M=v, lanes16-31 -> M=v+8, N=lane&15
// ---------------------------------------------------------------------------
template <int K, int N>
__global__ void __launch_bounds__(256) gemm_wmma_kernel(
    const float* __restrict__ A, const float* __restrict__ B,
    float* __restrict__ C, int M)
{
  const long tid  = (long)blockIdx.x * blockDim.x + threadIdx.x;
  const int  wave = (int)(tid >> 5);
  const int  lane = threadIdx.x & 31;
  constexpr int NT = N >> 5;                // 32-wide column tiles
  const int  mt = wave / NT;
  const int  nt = wave - mt * NT;
  const int  row0 = mt << 5;                // 32-row tile
  if (row0 >= M) return;                    // wave-uniform exit
  const bool full  = (row0 + 32 <= M);      // wave-uniform: remainder tile?
  const int  rowA1 = full ? (row0 + 16) : row0;  // clamp 2nd frag if remainder
  const int  col0 = nt << 5;
  const int  lh = lane >> 4;                // lane half: 0 or 1
  const int  li = lane & 15;

#if USE_WMMA_F32
  v8f acc00 = {0.f,0.f,0.f,0.f,0.f,0.f,0.f,0.f};
  v8f acc01 = acc00, acc10 = acc00, acc11 = acc00;
  const float* arow0 = A + (size_t)(row0  + li) * K;
  const float* arow1 = A + (size_t)(rowA1 + li) * K;
  const float* bcol  = B + col0 + li;
  for (int k = 0; k < K; k += 4) {
    const int ka = k + (lh << 1);
    v2f a0; a0.x = arow0[ka]; a0.y = arow0[ka + 1];   // contiguous -> b64
    v2f a1; a1.x = arow1[ka]; a1.y = arow1[ka + 1];
    const float* bp = bcol + (size_t)ka * N;          // N constant -> imm offsets
    v2f b0; b0.x = bp[0];  b0.y = bp[N];
    v2f b1; b1.x = bp[16]; b1.y = bp[N + 16];
    acc00 = __builtin_amdgcn_wmma_f32_16x16x4_f32(false, a0, false, b0,
                                                  (short)0, acc00, false, false);
    acc01 = __builtin_amdgcn_wmma_f32_16x16x4_f32(false, a0, false, b1,
                                                  (short)0, acc01, false, false);
    acc10 = __builtin_amdgcn_wmma_f32_16x16x4_f32(false, a1, false, b0,
                                                  (short)0, acc10, false, false);
    acc11 = __builtin_amdgcn_wmma_f32_16x16x4_f32(false, a1, false, b1,
                                                  (short)0, acc11, false, false);
  }
#pragma unroll
  for (int v = 0; v < 8; ++v) {
    const int r = v + (lh << 3);
    float* crow = C + (size_t)(row0 + r) * N + col0 + li;
    crow[0]  = acc00[v];
    crow[16] = acc01[v];
  }
  if (full) {
#pragma unroll
    for (int v = 0; v < 8; ++v) {
      const int r = 16 + v + (lh << 3);
      float* crow = C + (size_t)(row0 + r) * N + col0 + li;
      crow[0]  = acc10[v];
      crow[16] = acc11[v];
    }
  }
#else
  // Scalar fallback (also what the host pass sees): identical tile mapping.
  const int rows = full ? 32 : 16;
  for (int rv = 0; rv < rows; rv += 16) {
    for (int v = 0; v < 8; ++v) {
      const int r = row0 + rv + v + (lh << 3);
      const float* ar = A + (size_t)r * K;
      float s0 = 0.f, s1 = 0.f;
      for (int k = 0; k < K; ++k) {
        const float a = ar[k];
        s0 += a * B[(size_t)k * N + col0 + li];
        s1 += a * B[(size_t)k * N + col0 + 16 + li];
      }
      C[(size_t)r * N + col0 + li]      = s0;
      C[(size_t)r * N + col0 + 16 + li] = s1;
    }
  }
#endif
}

// ---------------------------------------------------------------------------
// Attention logits: al_s[n,h] = <h[n,h,:], a_src[h,:]>, al_d likewise.
// ---------------------------------------------------------------------------
__global__ void __launch_bounds__(256) compute_al_kernel(
    const float* __restrict__ h, const float* __restrict__ a_src,
    const float* __restrict__ a_dst, float* __restrict__ als,
    float* __restrict__ ald, int N, int H, int C)
{
  const int i = blockIdx.x * blockDim.x + threadIdx.x;
  if (i >= N * H) return;
  const int n = i / H, hh = i - n * H;
  const float* hp = h + (size_t)n * H * C + (size_t)hh * C;
  const float* as = a_src + hh * C;
  const float* ad = a_dst + hh * C;
  float ss = 0.f, sd = 0.f;
  for (int c = 0; c < C; ++c) { const float v = hp[c]; ss += v * as[c]; sd += v * ad[c]; }
  als[i] = ss; ald[i] = sd;
}

__global__ void __launch_bounds__(256) fill_kernel(float* __restrict__ p, float v, long n)
{
  const long i = (long)blockIdx.x * blockDim.x + threadIdx.x;
  if (i < n) p[i] = v;
}

__device__ __forceinline__ void edge_sd(const int* __restrict__ ei, int E, int e,
                                        int& s, int& d)
{
  if (e < E) { s = ei[e]; d = ei[E + e]; }
  else       { const int n = e - E; s = n; d = n; }   // appended self-loops
}

__device__ __forceinline__ float leaky(float v) { return v > 0.f ? v : 0.2f * v; }

__device__ __forceinline__ void atomicMaxF(float* addr, float val)
{
  int* ia = (int*)addr;
  int cur = __float_as_int(*addr);
  while (val > __int_as_float(cur)) {
    const int prev = atomicCAS(ia, cur, __float_as_int(val));
    if (prev == cur) break;
    cur = prev;
  }
}

// Pass 1: segment max of leaky_relu(al_s[src]+al_d[dst]) over dst.
__global__ void __launch_bounds__(256) edge_max_kernel(
    const int* __restrict__ ei, int E, int NE,
    const float* __restrict__ als, const float* __restrict__ ald,
    float* __restrict__ m, int H)
{
  const int e = blockIdx.x * blockDim.x + threadIdx.x;
  if (e >= NE) return;
  int s, d; edge_sd(ei, E, e, s, d);
  for (int h = 0; h < H; ++h)
    atomicMaxF(&m[(size_t)d * H + h], leaky(als[(size_t)s * H + h] + ald[(size_t)d * H + h]));
}

// Pass 2: ee = exp(e - m[dst]); den[dst] += ee (global_atomic_add_f32).
__global__ void __launch_bounds__(256) edge_expsum_kernel(
    const int* __restrict__ ei, int E, int NE,
    const float* __restrict__ als, const float* __restrict__ ald,
    const float* __restrict__ m, float* __restrict__ ee,
    float* __restrict__ den, int H)
{
  const int e = blockIdx.x * blockDim.x + threadIdx.x;
  if (e >= NE) return;
  int s, d; edge_sd(ei, E, e, s, d);
  for (int h = 0; h < H; ++h) {
    const float v  = leaky(als[(size_t)s * H + h] + ald[(size_t)d * H + h]);
    const float ex = __expf(v - m[(size_t)d * H + h]);
    ee[(size_t)e * H + h] = ex;
    atomicAdd(&den[(size_t)d * H + h], ex);
  }
}

__global__ void __launch_bounds__(256) recip_kernel(float* __restrict__ p, long n)
{
  const long i = (long)blockIdx.x * blockDim.x + threadIdx.x;
  if (i < n) p[i] = 1.0f / p[i];
}

// Pass 3: acc[dst, j] += h[src, j] * (ee[e,h] * rden[dst,h]);  j in [0,HC)
__global__ void __launch_bounds__(256) edge_aggr_kernel(
    const int* __restrict__ ei, int E, long total,
    const float* __restrict__ hsrc, const float* __restrict__ ee,
    const float* __restrict__ rden, float* __restrict__ acc,
    int H, int HC, int logC, int logHC)
{
  const long g = (long)blockIdx.x * blockDim.x + threadIdx.x;
  if (g >= total) return;
  const int  j = (int)(g & (long)(HC - 1));
  const int  e = (int)(g >> logHC);
  int s, d; edge_sd(ei, E, e, s, d);
  const int  h = j >> logC;
  const float alpha = ee[(size_t)e * H + h] * rden[(size_t)d * H + h];
  atomicAdd(&acc[(size_t)d * HC + j], hsrc[(size_t)s * HC + j] * alpha);
}

// Epilogues
__global__ void __launch_bounds__(256) post_elu_kernel(
    const float* __restrict__ acc, const float* __restrict__ b,
    float* __restrict__ out, long n, int mask)
{
  const long i = (long)blockIdx.x * blockDim.x + threadIdx.x;
  if (i >= n) return;
  const float v = acc[i] + b[i & mask];
  out[i] = v > 0.f ? v : expm1f(v);
}

__global__ void __launch_bounds__(256) post_elu_res_kernel(
    const float* __restrict__ acc, const float* __restrict__ b,
    const float* __restrict__ res, float* __restrict__ out, long n, int mask)
{
  const long i = (long)blockIdx.x * blockDim.x + threadIdx.x;
  if (i >= n) return;
  float v = acc[i] + b[i & mask];
  v = v > 0.f ? v : expm1f(v);
  out[i] = v + res[i];
}

__global__ void __launch_bounds__(256) post_bias_kernel(
    const float* __restrict__ acc, const float* __restrict__ b,
    float* __restrict__ out, long n, int mask)
{
  const long i = (long)blockIdx.x * blockDim.x + threadIdx.x;
  if (i >= n) return;
  out[i] = acc[i] + b[i & mask];
}

// ---------------------------------------------------------------------------
static inline int cdiv_l(long a, long b) { return (int)((a + b - 1) / b); }

extern "C" void kernel_launch(void* const* d_in, const int* in_sizes, int n_in,
                              void* d_out, int out_size, void* d_ws, size_t ws_size,
                              hipStream_t stream)
{
  const float* x       = (const float*)d_in[0];
  const int*   ei      = (const int*)  d_in[1];
  const float* W_in    = (const float*)d_in[2];
  const float* a_s_in  = (const float*)d_in[3];
  const float* a_d_in  = (const float*)d_in[4];
  const float* b_in    = (const float*)d_in[5];
  const float* W_mid   = (const float*)d_in[6];
  const float* a_s_mid = (const float*)d_in[7];
  const float* a_d_mid = (const float*)d_in[8];
  const float* b_mid   = (const float*)d_in[9];
  const float* W_out   = (const float*)d_in[10];
  const float* a_s_out = (const float*)d_in[11];
  const float* a_d_out = (const float*)d_in[12];
  const float* b_out   = (const float*)d_in[13];

  const int F = 512, H = 4, C = 32, HC = 128, CL = 64;
  const int N  = in_sizes[0] / F;      // 50000
  const int E  = in_sizes[1] / 2;      // 800000
  const int NE = E + N;                // + self-loops

  // Workspace layout (floats)
  float* bufA = (float*)d_ws;               // [N,HC] projections; later layer-2 output
  float* bufB = bufA + (size_t)N * HC;      // [N,HC] aggregation accumulator
  float* bufC = bufB + (size_t)N * HC;      // [N,HC] layer-1 activations; later proj3
  float* als  = bufC + (size_t)N * HC;      // [N,H]
  float* ald  = als  + (size_t)N * H;       // [N,H]
  float* mbuf = ald  + (size_t)N * H;       // [N,H]
  float* den  = mbuf + (size_t)N * H;       // [N,H]
  float* ee   = den  + (size_t)N * H;       // [NE,H]

  const int BLK = 256;
  const float NEG_INF = -3.0e38f;

  // ---------------- Layer 1: x[N,512] -> h1 = elu(agg + b_in) ----------------
  {
    const long waves = (long)((N + 31) / 32) * (HC / 32);
    gemm_wmma_kernel<512, 128><<<cdiv_l(waves * 32, BLK), BLK, 0, stream>>>(x, W_in, bufA, N);
    compute_al_kernel<<<cdiv_l((long)N * H, BLK), BLK, 0, stream>>>(bufA, a_s_in, a_d_in, als, ald, N, H, C);
    fill_kernel<<<cdiv_l((long)N * H, BLK), BLK, 0, stream>>>(mbuf, NEG_INF, (long)N * H);
    fill_kernel<<<cdiv_l((long)N * H, BLK), BLK, 0, stream>>>(den, 0.f, (long)N * H);
    fill_kernel<<<cdiv_l((long)N * HC, BLK), BLK, 0, stream>>>(bufB, 0.f, (long)N * HC);
    edge_max_kernel<<<cdiv_l(NE, BLK), BLK, 0, stream>>>(ei, E, NE, als, ald, mbuf, H);
    edge_expsum_kernel<<<cdiv_l(NE, BLK), BLK, 0, stream>>>(ei, E, NE, als, ald, mbuf, ee, den, H);
    recip_kernel<<<cdiv_l((long)N * H, BLK), BLK, 0, stream>>>(den, (long)N * H);
    edge_aggr_kernel<<<cdiv_l((long)NE * HC, BLK), BLK, 0, stream>>>(
        ei, E, (long)NE * HC, bufA, ee, den, bufB, H, HC, 5 /*log2 C*/, 7 /*log2 HC*/);
    post_elu_kernel<<<cdiv_l((long)N * HC, BLK), BLK, 0, stream>>>(bufB, b_in, bufC, (long)N * HC, HC - 1);
  }

  // ---------------- Layer 2: h2 = elu(agg + b_mid) + h1 (residual) -----------
  {
    const long waves = (long)((N + 31) / 32) * (HC / 32);
    gemm_wmma_kernel<128, 128><<<cdiv_l(waves * 32, BLK), BLK, 0, stream>>>(bufC, W_mid, bufA, N);
    compute_al_kernel<<<cdiv_l((long)N * H, BLK), BLK, 0, stream>>>(bufA, a_s_mid, a_d_mid, als, ald, N, H, C);
    fill_kernel<<<cdiv_l((long)N * H, BLK), BLK, 0, stream>>>(mbuf, NEG_INF, (long)N * H);
    fill_kernel<<<cdiv_l((long)N * H, BLK), BLK, 0, stream>>>(den, 0.f, (long)N * H);
    fill_kernel<<<cdiv_l((long)N * HC, BLK), BLK, 0, stream>>>(bufB, 0.f, (long)N * HC);
    edge_max_kernel<<<cdiv_l(NE, BLK), BLK, 0, stream>>>(ei, E, NE, als, ald, mbuf, H);
    edge_expsum_kernel<<<cdiv_l(NE, BLK), BLK, 0, stream>>>(ei, E, NE, als, ald, mbuf, ee, den, H);
    recip_kernel<<<cdiv_l((long)N * H, BLK), BLK, 0, stream>>>(den, (long)N * H);
    edge_aggr_kernel<<<cdiv_l((long)NE * HC, BLK), BLK, 0, stream>>>(
        ei, E, (long)NE * HC, bufA, ee, den, bufB, H, HC, 5, 7);
    // bufA (proj2) no longer needed -> becomes layer-2 output
    post_elu_res_kernel<<<cdiv_l((long)N * HC, BLK), BLK, 0, stream>>>(
        bufB, b_mid, bufC, bufA, (long)N * HC, HC - 1);
  }

  // ---------------- Layer 3: H=1, C=64; out = agg + b_out --------------------
  {
    const long waves = (long)((N + 31) / 32) * (CL / 32);
    gemm_wmma_kernel<128, 64><<<cdiv_l(waves * 32, BLK), BLK, 0, stream>>>(bufA, W_out, bufC, N);
    compute_al_kernel<<<cdiv_l((long)N, BLK), BLK, 0, stream>>>(bufC, a_s_out, a_d_out, als, ald, N, 1, CL);
    fill_kernel<<<cdiv_l((long)N, BLK), BLK, 0, stream>>>(mbuf, NEG_INF, (long)N);
    fill_kernel<<<cdiv_l((long)N, BLK), BLK, 0, stream>>>(den, 0.f, (long)N);
    fill_kernel<<<cdiv_l((long)N * CL, BLK), BLK, 0, stream>>>(bufB, 0.f, (long)N * CL);
    edge_max_kernel<<<cdiv_l(NE, BLK), BLK, 0, stream>>>(ei, E, NE, als, ald, mbuf, 1);
    edge_expsum_kernel<<<cdiv_l(NE, BLK), BLK, 0, stream>>>(ei, E, NE, als, ald, mbuf, ee, den, 1);
    recip_kernel<<<cdiv_l((long)N, BLK), BLK, 0, stream>>>(den, (long)N);
    edge_aggr_kernel<<<cdiv_l((long)NE * CL, BLK), BLK, 0, stream>>>(
        ei, E, (long)NE * CL, bufC, ee, den, bufB, 1, CL, 6 /*log2 64*/, 6);
    post_bias_kernel<<<cdiv_l((long)N * CL, BLK), BLK, 0, stream>>>(
        bufB, b_out, (float*)d_out, (long)N * CL, CL - 1);
  }
}